// VectorNet_29506425323801
// MI455X (gfx1250) — compile-verified
//
#include <hip/hip_runtime.h>
#include <math.h>

typedef __attribute__((ext_vector_type(2))) float v2f;
typedef __attribute__((ext_vector_type(8))) float v8f;

#define HD 128
#define LL 32
#define NEGBIG -10000.0f

__device__ __forceinline__ v8f wmma4(v2f a, v2f b, v8f c) {
  // V_WMMA_F32_16X16X4_F32 : D = A(16x4) * B(4x16) + C(16x16), fp32
  return __builtin_amdgcn_wmma_f32_16x16x4_f32(false, a, false, b, (short)0, c,
                                               false, false);
}

__device__ __forceinline__ float wave_sum(float v) {
  #pragma unroll
  for (int o = 16; o; o >>= 1) v += __shfl_xor(v, o, 32);
  return v;
}
__device__ __forceinline__ float wave_max(float v) {
  #pragma unroll
  for (int o = 16; o; o >>= 1) v = fmaxf(v, __shfl_xor(v, o, 32));
  return v;
}

// out[M x N](ldo) (+)= A[M x K](lda) @ B[K x N](ldb) + bias
// A, B, out may be LDS or global pointers. M,N mult of 16; K mult of 4.
__device__ void gemm_AB(const float* __restrict__ A, int lda,
                        const float* __restrict__ B, int ldb,
                        const float* __restrict__ bias,
                        float* __restrict__ out, int ldo,
                        int M, int N, int K, bool accum, int tid, int nwaves) {
  const int wave = tid >> 5, lane = tid & 31;
  const int half = lane >> 4, i16 = lane & 15;
  const int mt = M >> 4, nt = N >> 4, ntiles = mt * nt;
  for (int t = wave; t < ntiles; t += nwaves) {
    const int tm = (t % mt) << 4, tn = (t / mt) << 4;
    v8f acc = {0.f, 0.f, 0.f, 0.f, 0.f, 0.f, 0.f, 0.f};
    const float* arow = A + (size_t)(tm + i16) * lda + 2 * half;
    const float* bcol = B + (size_t)(2 * half) * ldb + tn + i16;
    for (int k = 0; k < K; k += 4) {
      v2f a, b;
      a.x = arow[k];
      a.y = arow[k + 1];
      b.x = bcol[(size_t)k * ldb];
      b.y = bcol[(size_t)(k + 1) * ldb];
      acc = wmma4(a, b, acc);
    }
    const float bb = bias ? bias[tn + i16] : 0.f;
    float* o = out + (size_t)(tm + 8 * half) * ldo + tn + i16;
    #pragma unroll
    for (int r = 0; r < 8; ++r) {
      float v = acc[r] + bb;
      if (accum) o[(size_t)r * ldo] += v;
      else       o[(size_t)r * ldo] = v;
    }
  }
}

// out[M x N](ldo) = A[M x K](lda) @ Bm^T   where Bm is [N x K](ldb) row-major
__device__ void gemm_ABt(const float* __restrict__ A, int lda,
                         const float* __restrict__ Bm, int ldb,
                         float* __restrict__ out, int ldo,
                         int M, int N, int K, int tid, int nwaves) {
  const int wave = tid >> 5, lane = tid & 31;
  const int half = lane >> 4, i16 = lane & 15;
  const int mt = M >> 4, nt = N >> 4, ntiles = mt * nt;
  for (int t = wave; t < ntiles; t += nwaves) {
    const int tm = (t % mt) << 4, tn = (t / mt) << 4;
    v8f acc = {0.f, 0.f, 0.f, 0.f, 0.f, 0.f, 0.f, 0.f};
    const float* arow = A + (size_t)(tm + i16) * lda + 2 * half;
    const float* brow = Bm + (size_t)(tn + i16) * ldb + 2 * half;
    for (int k = 0; k < K; k += 4) {
      v2f a, b;
      a.x = arow[k];
      a.y = arow[k + 1];
      b.x = brow[k];
      b.y = brow[k + 1];
      acc = wmma4(a, b, acc);
    }
    float* o = out + (size_t)(tm + 8 * half) * ldo + tn + i16;
    #pragma unroll
    for (int r = 0; r < 8; ++r) o[(size_t)r * ldo] = acc[r];
  }
}

// ---------------- Kernel 1: per-polyline encoder ----------------

// LayerNorm over 32 rows x 128 cols (4 waves: wave handles rows wave,wave+4,...)
// pre_relu_add: x = relu(buf)+resid before LN ; post_relu: relu after LN.
__device__ void ln_rows32(float* buf, const float* resid, const float* g,
                          const float* be, bool pre_relu_add, bool post_relu,
                          int tid) {
  const int wave = tid >> 5, lane = tid & 31;
  for (int r = wave; r < 32; r += 4) {
    float x[4];
    float s = 0.f;
    #pragma unroll
    for (int j = 0; j < 4; ++j) {
      const int c = lane + 32 * j;
      float v = buf[r * HD + c];
      if (pre_relu_add) v = fmaxf(v, 0.f) + resid[r * HD + c];
      x[j] = v;
      s += v;
    }
    s = wave_sum(s);
    const float mu = s * (1.f / 128.f);
    float vs = 0.f;
    #pragma unroll
    for (int j = 0; j < 4; ++j) {
      const float d = x[j] - mu;
      vs += d * d;
    }
    vs = wave_sum(vs) * (1.f / 128.f);
    const float inv = rsqrtf(vs + 1e-12f);
    #pragma unroll
    for (int j = 0; j < 4; ++j) {
      const int c = lane + 32 * j;
      float y = (x[j] - mu) * inv * g[c] + be[c];
      if (post_relu) y = fmaxf(y, 0.f);
      buf[r * HD + c] = y;
    }
  }
}

// masked softmax over 2 heads x 32 rows x 32 cols, scale = 1/sqrt(64)
__device__ void softmax_masked(float* ss, int len, int tid) {
  const int wave = tid >> 5, lane = tid & 31;
  for (int rr = wave; rr < 64; rr += 4) {
    const int head = rr >> 5, i = rr & 31;
    float* row = ss + head * 1024 + i * 32;
    float v = row[lane] * 0.125f + ((i < len && lane < len) ? 0.f : NEGBIG);
    const float mx = wave_max(v);
    const float e = __expf(v - mx);
    const float sm = wave_sum(e);
    row[lane] = e / sm;
  }
}

__global__ void __launch_bounds__(128) polyline_kernel(
    const float* __restrict__ vectors, const int* __restrict__ lengths,
    const float* __restrict__ w0, const float* __restrict__ b0,
    const float* __restrict__ g0, const float* __restrict__ be0,
    const float* __restrict__ qkvw, const float* __restrict__ qkvb,
    const float* __restrict__ lng, const float* __restrict__ lnb,
    float* __restrict__ poly) {
  __shared__ float sh[LL * HD];   // activations h
  __shared__ float st[LL * HD];   // x / residual tmp
  __shared__ float sq[LL * HD];
  __shared__ float sk[LL * HD];
  __shared__ float sv[LL * HD];
  __shared__ float ss[2 * LL * LL];  // scores, 2 heads

  const int bp = blockIdx.x;
  const int tid = threadIdx.x;
  int len = lengths[bp];
  len = min(max(len, 1), LL);

  // masked input -> st
  const float* xg = vectors + (size_t)bp * (LL * HD);
  for (int i = tid; i < LL * HD; i += 128) {
    const int row = i >> 7;
    st[i] = (row < len) ? xg[i] : 0.f;
  }
  __syncthreads();

  // h = relu(LN(x @ w0 + b0))
  gemm_AB(st, HD, w0, HD, b0, sh, HD, LL, HD, HD, false, tid, 4);
  __syncthreads();
  ln_rows32(sh, nullptr, g0, be0, false, true, tid);
  __syncthreads();

  for (int d = 0; d < 3; ++d) {
    const float* W = qkvw + (size_t)d * 3 * HD * HD;
    const float* Bi = qkvb + (size_t)d * 3 * HD;
    // tmp = h
    for (int i = tid; i < LL * HD; i += 128) st[i] = sh[i];
    // q,k,v
    gemm_AB(sh, HD, W,               HD, Bi,           sq, HD, LL, HD, HD, false, tid, 4);
    gemm_AB(sh, HD, W + HD * HD,     HD, Bi + HD,      sk, HD, LL, HD, HD, false, tid, 4);
    gemm_AB(sh, HD, W + 2 * HD * HD, HD, Bi + 2 * HD,  sv, HD, LL, HD, HD, false, tid, 4);
    __syncthreads();
    // per-head scores S = Q_h @ K_h^T  (hs = 64)
    for (int hh = 0; hh < 2; ++hh)
      gemm_ABt(sq + hh * 64, HD, sk + hh * 64, HD, ss + hh * 1024, 32,
               LL, LL, 64, tid, 4);
    __syncthreads();
    softmax_masked(ss, len, tid);
    __syncthreads();
    // O_h = P_h @ V_h  -> overwrite sh
    for (int hh = 0; hh < 2; ++hh)
      gemm_AB(ss + hh * 1024, 32, sv + hh * 64, HD, nullptr,
              sh + hh * 64, HD, LL, 64, LL, false, tid, 4);
    __syncthreads();
    // h = LN(relu(o) + tmp)
    ln_rows32(sh, st, lng + d * HD, lnb + d * HD, true, false, tid);
    __syncthreads();
  }

  // poly = max over L rows
  float m = sh[tid];
  for (int r = 1; r < LL; ++r) m = fmaxf(m, sh[r * HD + tid]);
  poly[(size_t)bp * HD + tid] = m;
}

// ---------------- Kernel 2: cross attentions (one block per batch) ----------

__device__ void softmax_rows(float* s, int rows, int cols, int lds, float scale,
                             int tid, int nwaves) {
  const int wave = tid >> 5, lane = tid & 31;
  for (int r = wave; r < rows; r += nwaves) {
    float v[4];
    float mx = -3.0e38f;
    #pragma unroll
    for (int j = 0; j < 4; ++j) {
      const int c = lane + 32 * j;
      v[j] = (c < cols) ? s[(size_t)r * lds + c] * scale : -3.0e38f;
      mx = fmaxf(mx, v[j]);
    }
    mx = wave_max(mx);
    float sm = 0.f;
    #pragma unroll
    for (int j = 0; j < 4; ++j) {
      const int c = lane + 32 * j;
      if (c < cols) { v[j] = __expf(v[j] - mx); sm += v[j]; }
    }
    sm = wave_sum(sm);
    const float inv = 1.f / sm;
    #pragma unroll
    for (int j = 0; j < 4; ++j) {
      const int c = lane + 32 * j;
      if (c < cols) s[(size_t)r * lds + c] = v[j] * inv;
    }
  }
}

// single-head MHA: out[qrows x hs](ld 128) (+)= softmax(qWq (kWk)^T/sqrt(hs)) (vWv)
__device__ void mha_step(const float* qin, int qrows, const float* kvin,
                         int kvrows, const float* W, const float* Bi, int hs,
                         float* outp, bool accum, float* qbuf, float* skv,
                         float* ssc, int tid) {
  // q -> global scratch, k -> LDS
  gemm_AB(qin, HD, W, hs, Bi, qbuf, hs, qrows, hs, HD, false, tid, 8);
  gemm_AB(kvin, HD, W + HD * hs, hs, Bi + hs, skv, hs, kvrows, hs, HD, false,
          tid, 8);
  __threadfence_block();
  __syncthreads();
  // scores = q @ k^T  (ld 128)
  gemm_ABt(qbuf, hs, skv, hs, ssc, HD, qrows, kvrows, hs, tid, 8);
  __syncthreads();
  // v -> LDS (overwrite k) ; softmax on scores
  gemm_AB(kvin, HD, W + 2 * HD * hs, hs, Bi + 2 * hs, skv, hs, kvrows, hs, HD,
          false, tid, 8);
  softmax_rows(ssc, qrows, kvrows, HD, rsqrtf((float)hs), tid, 8);
  __syncthreads();
  // o = p @ v -> out (global)
  gemm_AB(ssc, HD, skv, hs, nullptr, outp, HD, qrows, hs, kvrows, accum, tid, 8);
  __threadfence_block();
  __syncthreads();
}

__global__ void __launch_bounds__(256) cross_kernel(
    float* __restrict__ poly, float* __restrict__ qscr,
    float* __restrict__ delta, float* __restrict__ out,
    const float* __restrict__ a2lw, const float* __restrict__ a2lb,
    const float* __restrict__ l2lw, const float* __restrict__ l2lb,
    const float* __restrict__ l2aw, const float* __restrict__ l2ab,
    const float* __restrict__ ggw, const float* __restrict__ ggb) {
  __shared__ float skv[HD * HD];  // K then V staging
  __shared__ float ssc[HD * HD];  // scores
  const int b = blockIdx.x;
  const int tid = threadIdx.x;
  float* P = poly + (size_t)b * HD * HD;    // [128][128], rows 0..15 agents
  float* lanes = P + 16 * HD;               // 112 rows
  float* qb = qscr + (size_t)b * HD * HD;
  float* dl = delta + (size_t)b * 112 * HD;

  // lanes += mha(lanes, agents)           hs=128
  mha_step(lanes, 112, P, 16, a2lw, a2lb, HD, lanes, true, qb, skv, ssc, tid);
  // delta = concat(mha(lanes,lanes,l2l0), mha(lanes,lanes,l2l1))  hs=64
  mha_step(lanes, 112, lanes, 112, l2lw, l2lb, 64, dl, false, qb, skv, ssc, tid);
  mha_step(lanes, 112, lanes, 112, l2lw + 3 * HD * 64, l2lb + 3 * 64, 64,
           dl + 64, false, qb, skv, ssc, tid);
  for (int i = tid; i < 112 * HD; i += 256) lanes[i] += dl[i];
  __threadfence_block();
  __syncthreads();
  // agents += mha(agents, lanes)          hs=128
  mha_step(P, 16, lanes, 112, l2aw, l2ab, HD, P, true, qb, skv, ssc, tid);
  // out = mha(poly, poly, gg)             hs=128
  mha_step(P, HD, P, HD, ggw, ggb, HD, out + (size_t)b * HD * HD, false, qb,
           skv, ssc, tid);
}

// ---------------- launch ----------------

extern "C" void kernel_launch(void* const* d_in, const int* in_sizes, int n_in,
                              void* d_out, int out_size, void* d_ws,
                              size_t ws_size, hipStream_t stream) {
  const float* vectors = (const float*)d_in[0];
  const int* lengths   = (const int*)d_in[1];
  // d_in[2] = map_start (16, fixed by problem setup)
  const float* sub0_w  = (const float*)d_in[3];
  const float* sub0_b  = (const float*)d_in[4];
  const float* ln0g    = (const float*)d_in[5];
  const float* ln0b    = (const float*)d_in[6];
  const float* qkvw    = (const float*)d_in[7];
  const float* qkvb    = (const float*)d_in[8];
  const float* lng     = (const float*)d_in[9];
  const float* lnb     = (const float*)d_in[10];
  const float* a2lw    = (const float*)d_in[11];
  const float* a2lb    = (const float*)d_in[12];
  const float* l2lw    = (const float*)d_in[13];
  const float* l2lb    = (const float*)d_in[14];
  const float* l2aw    = (const float*)d_in[15];
  const float* l2ab    = (const float*)d_in[16];
  const float* ggw     = (const float*)d_in[17];
  const float* ggb     = (const float*)d_in[18];

  float* out = (float*)d_out;
  float* ws = (float*)d_ws;
  const size_t BPH = (size_t)32 * 128 * 128;  // 524288
  float* poly  = ws;               // [B][128][128]
  float* qscr  = ws + BPH;         // [B][128][128] q scratch
  float* delta = ws + 2 * BPH;     // [B][112][128]

  polyline_kernel<<<32 * 128, 128, 0, stream>>>(vectors, lengths, sub0_w,
                                                sub0_b, ln0g, ln0b, qkvw, qkvb,
                                                lng, lnb, poly);
  cross_kernel<<<32, 256, 0, stream>>>(poly, qscr, delta, out, a2lw, a2lb,
                                       l2lw, l2lb, l2aw, l2ab, ggw, ggb);
}